// IPLS_3762391351865
// MI455X (gfx1250) — compile-verified
//
#include <hip/hip_runtime.h>
#include <math.h>

// ---------------------------------------------------------------------------
// Incremental PLS scan, persistent single-workgroup kernel for MI455X/gfx1250.
//  - All recurrent state (Wz 16x2048, P 16x2048, Cz 16x256, means, scalars)
//    resident in the 320KB WGP LDS for the whole 1024-step scan.
//  - Next step's (x,y) row prefetched with global_load_async_to_lds_* and
//    synchronized with s_wait_asynccnt (ASYNCcnt) -> overlap load w/ compute.
//  - Burn-phase Wz@x matvec done with v_wmma_f32_16x16x4_f32.
//  - wave32 shfl_xor reductions + cross-wave LDS combine (8 waves).
// ---------------------------------------------------------------------------

#define T_STEPS   1024
#define F_DIM     2048
#define G_DIM     256
#define NLAT      16
#define NTHREADS  256
#define FPT       (F_DIM / NTHREADS)   /* 8 elements of x per thread */
#define EPS_F     1e-7f
#define BURN_N    3
#define INNER     2

typedef __attribute__((ext_vector_type(2))) float v2f;
typedef __attribute__((ext_vector_type(8))) float v8f;

// LDS partition (floats)
#define OFF_WZ    0
#define OFF_P     (OFF_WZ  + NLAT * F_DIM)        // 32768
#define OFF_CZ    (OFF_P   + NLAT * F_DIM)        // 65536
#define OFF_MUX   (OFF_CZ  + NLAT * G_DIM)        // 69632
#define OFF_MUY   (OFF_MUX + F_DIM)               // 71680
#define OFF_XL    (OFF_MUY + G_DIM)               // 71936
#define OFF_XRAW  (OFF_XL  + F_DIM)               // 73984 (2 buffers)
#define OFF_YRAW  (OFF_XRAW + 2 * F_DIM)          // 78080 (2 buffers)
#define OFF_U     (OFF_YRAW + 2 * G_DIM)          // 78592
#define OFF_TSS   (OFF_U   + NLAT)
#define OFF_BZ    (OFF_TSS + NLAT)
#define OFF_WW    (OFF_BZ  + NLAT)
#define OFF_CC    (OFF_WW  + NLAT)
#define OFF_PP    (OFF_CC  + NLAT)
#define OFF_A16   (OFF_PP  + NLAT)
#define OFF_RED   (OFF_A16 + NLAT)                // 36 floats of scratch
#define LDS_FLOATS (OFF_RED + 40)
// total ~ 315KB  <  320KB WGP LDS

__device__ static inline unsigned lds_addr32(const void* p) {
  // LDS flat aperture keeps the byte offset in the low 32 bits.
  return (unsigned)(unsigned long long)p;
}

__device__ static inline void async_prefetch_row(const float* Xg, const float* Yg,
                                                 int s, float* xraw, float* yraw,
                                                 int tid) {
  unsigned long long xb = (unsigned long long)Xg + (unsigned long long)s * (F_DIM * 4ull);
  unsigned long long yb = (unsigned long long)Yg + (unsigned long long)s * (G_DIM * 4ull);
  unsigned vx0 = (unsigned)tid * 16u;                 // bytes [0, 4096)
  unsigned vx1 = vx0 + (unsigned)(F_DIM * 2);         // bytes [4096, 8192)
  unsigned vy  = (unsigned)tid * 4u;                  // bytes [0, 1024)
  unsigned lx0 = lds_addr32(xraw) + vx0;
  unsigned lx1 = lds_addr32(xraw) + vx1;
  unsigned ly  = lds_addr32(yraw) + vy;
  asm volatile("global_load_async_to_lds_b128 %0, %1, %2"
               :: "v"(lx0), "v"(vx0), "s"(xb) : "memory");
  asm volatile("global_load_async_to_lds_b128 %0, %1, %2"
               :: "v"(lx1), "v"(vx1), "s"(xb) : "memory");
  asm volatile("global_load_async_to_lds_b32 %0, %1, %2"
               :: "v"(ly), "v"(vy), "s"(yb) : "memory");
}

// Reduce four values across all 256 threads (8 wave32 waves).
__device__ static inline void reduce4(float a, float b, float c, float d,
                                      float* RED, int tid, float out[4]) {
  #pragma unroll
  for (int m = 16; m; m >>= 1) {
    a += __shfl_xor(a, m);
    b += __shfl_xor(b, m);
    c += __shfl_xor(c, m);
    d += __shfl_xor(d, m);
  }
  int w = tid >> 5;
  if ((tid & 31) == 0) {
    RED[w * 4 + 0] = a; RED[w * 4 + 1] = b;
    RED[w * 4 + 2] = c; RED[w * 4 + 3] = d;
  }
  __syncthreads();
  if (tid == 0) {
    float s0 = 0.f, s1 = 0.f, s2 = 0.f, s3 = 0.f;
    #pragma unroll
    for (int i = 0; i < 8; ++i) {
      s0 += RED[i * 4 + 0]; s1 += RED[i * 4 + 1];
      s2 += RED[i * 4 + 2]; s3 += RED[i * 4 + 3];
    }
    RED[32] = s0; RED[33] = s1; RED[34] = s2; RED[35] = s3;
  }
  __syncthreads();
  out[0] = RED[32]; out[1] = RED[33]; out[2] = RED[34]; out[3] = RED[35];
}

__global__ __launch_bounds__(NTHREADS, 1)
void ipls_persistent(const float* __restrict__ Xg, const float* __restrict__ Yg,
                     const float* __restrict__ mux_g, const float* __restrict__ muy_g,
                     const float* __restrict__ u_g, const float* __restrict__ wz_g,
                     const float* __restrict__ cz_g, const float* __restrict__ tss_g,
                     const float* __restrict__ bz_g, const float* __restrict__ p_g,
                     const int* __restrict__ n0_g, float* __restrict__ out) {
  extern __shared__ float SM[];
  float* WZ   = SM + OFF_WZ;
  float* PL   = SM + OFF_P;
  float* CZ   = SM + OFF_CZ;
  float* MUX  = SM + OFF_MUX;
  float* MUY  = SM + OFF_MUY;
  float* XL   = SM + OFF_XL;
  float* XRAW = SM + OFF_XRAW;
  float* YRAW = SM + OFF_YRAW;
  float* U    = SM + OFF_U;
  float* TSS  = SM + OFF_TSS;
  float* BZ   = SM + OFF_BZ;
  float* WWs  = SM + OFF_WW;
  float* CCs  = SM + OFF_CC;
  float* PPs  = SM + OFF_PP;
  float* A16  = SM + OFF_A16;
  float* RED  = SM + OFF_RED;

  const int tid  = threadIdx.x;
  const int g    = tid;                 // each thread owns one y element
  const int n0   = n0_g[0];

  // Kick off the first row load immediately (overlaps with state init).
  async_prefetch_row(Xg, Yg, 0, XRAW, YRAW, tid);

  // ---- load persistent state into LDS ----
  #pragma unroll
  for (int k = 0; k < FPT; ++k) {
    int j = tid + NTHREADS * k;
    MUX[j] = mux_g[j];
    for (int i = 0; i < NLAT; ++i) {
      WZ[i * F_DIM + j] = wz_g[i * F_DIM + j];
      PL[i * F_DIM + j] = p_g[i * F_DIM + j];
    }
  }
  MUY[g] = muy_g[g];
  for (int i = 0; i < NLAT; ++i) CZ[i * G_DIM + g] = cz_g[i * G_DIM + g];
  if (tid < NLAT) {
    U[tid]   = u_g[tid];
    TSS[tid] = tss_g[tid];
    BZ[tid]  = bz_g[tid];
  }
  __syncthreads();

  // ---- scalar invariants  WW=||Wz_i||^2, PP=||P_i||^2, CC=||Cz_i||^2 ----
  for (int i = 0; i < NLAT; ++i) {
    float pw = 0.f, pp = 0.f;
    #pragma unroll
    for (int k = 0; k < FPT; ++k) {
      int j = tid + NTHREADS * k;
      float wv = WZ[i * F_DIM + j], pv = PL[i * F_DIM + j];
      pw += wv * wv; pp += pv * pv;
    }
    float cv = CZ[i * G_DIM + g];
    float o[4];
    reduce4(pw, pp, cv * cv, 0.f, RED, tid, o);
    if (tid == 0) { WWs[i] = o[0]; PPs[i] = o[1]; CCs[i] = o[2]; }
  }
  __syncthreads();

  // ======================= sequential scan over T steps ====================
  for (int s = 0; s < T_STEPS; ++s) {
    const int buf = s & 1;

    // Wait for this step's async row, make it visible to all waves.
    asm volatile("s_wait_asynccnt 0" ::: "memory");
    __syncthreads();

    const float nf = (float)(n0 + s + 1);
    const float rc = 1.0f / (nf + 1.0f);

    // ---- update means, center, stash x in LDS (for WMMA), keep x in regs --
    float xr[FPT];
    #pragma unroll
    for (int k = 0; k < FPT; ++k) {
      int j = tid + NTHREADS * k;
      float x0 = XRAW[buf * F_DIM + j];
      float m  = (MUX[j] * nf + x0) * rc;
      MUX[j] = m;
      xr[k]  = x0 - m;
      XL[j]  = xr[k];
    }
    float y0 = YRAW[buf * G_DIM + g];
    float my = (MUY[g] * nf + y0) * rc;
    MUY[g] = my;
    float yr = y0 - my;

    // ---- xx = x.x, yy = y.y (fresh every step) ----
    float px = 0.f;
    #pragma unroll
    for (int k = 0; k < FPT; ++k) px += xr[k] * xr[k];
    float o4[4];
    reduce4(px, yr * yr, 0.f, 0.f, RED, tid, o4);
    float xx = o4[0], yy = o4[1];

    // Prefetch next row into the other buffer (all threads past centering
    // due to the barriers inside reduce4 above).
    if (s + 1 < T_STEPS)
      async_prefetch_row(Xg, Yg, s + 1,
                         XRAW + ((buf ^ 1) * F_DIM), YRAW + ((buf ^ 1) * G_DIM), tid);

    if ((n0 + s + 1) <= BURN_N) {
      // ================= burn-in: a16 = Wz @ x  via f32 WMMA ==============
      if (tid < NLAT) A16[tid] = 0.f;
      __syncthreads();

      const int lane = tid & 31, w = tid >> 5;
      const int row  = lane & 15;
      const int kh   = (lane >> 4) << 1;          // 0 for lanes 0-15, 2 for 16-31
      const float* wrow = &WZ[row * F_DIM];
      v8f acc = {};
      for (int it = 0; it < (F_DIM / 4) / 8; ++it) {   // 64 k-chunks per wave
        int kb = (w * 64 + it) * 4 + kh;
        v2f a = { wrow[kb], wrow[kb + 1] };       // A: 16x4 tile of Wz
        v2f b = { XL[kb],   XL[kb + 1] };         // B: x chunk broadcast to 16 cols
        acc = __builtin_amdgcn_wmma_f32_16x16x4_f32(
            false, a, false, b, (short)0, acc, false, false);
      }
      if ((lane & 15) == 0) {                     // lanes 0 and 16: column N=0
        float tc[8];
        #pragma unroll
        for (int r = 0; r < 8; ++r) tc[r] = acc[r];
        int mb = (lane >> 4) * 8;
        #pragma unroll
        for (int r = 0; r < 8; ++r) atomicAdd(&A16[mb + r], tc[r]);
      }
      __syncthreads();

      for (int i = 0; i < NLAT; ++i) {
        float ai  = A16[i];
        float ui  = U[i];
        float num = ai + ui * xx;                        // x . (Wz_i + ui*x)
        float ww2 = WWs[i] + 2.f * ui * ai + ui * ui * xx;
        float tz  = num / (sqrtf(ww2) + EPS_F);
        #pragma unroll
        for (int k = 0; k < FPT; ++k) {
          int j = tid + NTHREADS * k;
          WZ[i * F_DIM + j] += xr[k] * ui;
        }
        __syncthreads();
        if (tid == 0) { WWs[i] = ww2; TSS[i] += tz * tz; }
      }
    } else {
      // ======================== main: 16-latent deflation =================
      for (int i = 0; i < NLAT; ++i) {
        const float ui0 = U[i], tssi = TSS[i], bzi = BZ[i];
        const float wwi = WWs[i], cci = CCs[i], ppi = PPs[i];

        // per-latent reductions: xW = x.Wz_i, xP = x.P_i, yC = y.Cz_i
        float pxW = 0.f, pxP = 0.f;
        #pragma unroll
        for (int k = 0; k < FPT; ++k) {
          int j = tid + NTHREADS * k;
          pxW += xr[k] * WZ[i * F_DIM + j];
          pxP += xr[k] * PL[i * F_DIM + j];
        }
        float pyC = yr * CZ[i * G_DIM + g];
        reduce4(pxW, pxP, pyC, 0.f, RED, tid, o4);
        const float xW = o4[0], xP = o4[1], yC = o4[2];

        // inner scalar recurrence (hoisted dot products)
        float ui = ui0, uiW = ui0, tz = 0.f, tssn = tssi, t = 0.f, cc2 = cci;
        #pragma unroll
        for (int it = 0; it < INNER; ++it) {
          uiW = ui;
          float xwz = xW + ui * xx;
          float ww2 = wwi + 2.f * ui * xW + ui * ui * xx;
          tz   = xwz / (sqrtf(ww2) + EPS_F);
          tssn = tssi + tz * tz;
          t    = tz / sqrtf(tssn);
          float ycz = yC + t * yy;
          cc2 = cci + 2.f * t * yC + t * t * yy;
          ui  = ycz / sqrtf(cc2);
        }
        const float bznew = bzi + ui * tz;
        const float b     = bznew / sqrtf(tssn);
        const float icc   = 1.0f / sqrtf(cc2);
        const float bt    = b * t;

        // vector updates on thread-owned elements (no cross-thread deps)
        #pragma unroll
        for (int k = 0; k < FPT; ++k) {
          int j = tid + NTHREADS * k;
          WZ[i * F_DIM + j] += xr[k] * uiW;
          float pn = PL[i * F_DIM + j] + xr[k] * t;
          PL[i * F_DIM + j] = pn;
          xr[k] -= t * pn;                              // deflate x
        }
        float czn = CZ[i * G_DIM + g] + yr * t;
        CZ[i * G_DIM + g] = czn;
        yr -= bt * czn * icc;                           // deflate y

        // scalar-state updates (identical on all threads; thread 0 commits)
        const float wwn = wwi + 2.f * uiW * xW + uiW * uiW * xx;
        const float ppn = ppi + 2.f * t * xP + t * t * xx;
        const float xPn = xP + t * xx;                  // x . P_i_new
        xx = xx - 2.f * t * xPn + t * t * ppn;
        yy = yy - 2.f * bt * ui + bt * bt;
        if (tid == 0) {
          U[i] = ui; TSS[i] = tssn; BZ[i] = bznew;
          WWs[i] = wwn; CCs[i] = cc2; PPs[i] = ppn;
        }
      }
    }
    __syncthreads();
  }

  // ---- write final P (16 x 2048) ----
  for (int idx = tid; idx < NLAT * F_DIM; idx += NTHREADS) out[idx] = PL[idx];
}

extern "C" void kernel_launch(void* const* d_in, const int* in_sizes, int n_in,
                              void* d_out, int out_size, void* d_ws, size_t ws_size,
                              hipStream_t stream) {
  (void)in_sizes; (void)n_in; (void)out_size; (void)d_ws; (void)ws_size;
  const float* X    = (const float*)d_in[0];
  const float* Y    = (const float*)d_in[1];
  const float* mux  = (const float*)d_in[2];
  const float* muy  = (const float*)d_in[3];
  const float* u    = (const float*)d_in[4];
  const float* Wz   = (const float*)d_in[5];
  const float* Cz   = (const float*)d_in[6];
  const float* tss  = (const float*)d_in[7];
  const float* bz   = (const float*)d_in[8];
  const float* P    = (const float*)d_in[9];
  const int*   n0   = (const int*)d_in[10];
  float* out = (float*)d_out;

  const size_t lds_bytes = (size_t)LDS_FLOATS * sizeof(float);
  hipLaunchKernelGGL(ipls_persistent, dim3(1), dim3(NTHREADS), lds_bytes, stream,
                     X, Y, mux, muy, u, Wz, Cz, tss, bz, P, n0, out);
}